// CTC_34892314313734
// MI455X (gfx1250) — compile-verified
//
#include <hip/hip_runtime.h>
#include <hip/hip_bf16.h>
#include <cstddef>
#include <cstdint>

typedef __attribute__((ext_vector_type(2))) float v2f;
typedef __attribute__((ext_vector_type(8))) float v8f;
typedef __attribute__((ext_vector_type(4))) unsigned int v4u;
typedef __attribute__((ext_vector_type(4))) int v4i;
typedef __attribute__((ext_vector_type(8))) int v8i;

#define CTC_B   16
#define CTC_T   1000
#define CTC_D   512
#define CTC_V   5000
#define CTC_L   100
#define CTC_S   201            // 2L+1
#define CTC_J   (CTC_L + 1)    // blank + L labels
#define NEGV    (-1e30f)

// ---------------------------------------------------------------------------
// Kernel A: fused projection GEMM (f32 WMMA 16x16x4) + per-row online log-sum-exp.
// One block = 16 rows of the flattened (B*T, D) activation matrix.
// A tile staged into LDS by the Tensor Data Mover (one TENSOR_LOAD_TO_LDS,
// hardware-padded +2 dwords per 256 -> conflict-free 516-float row stride).
// 8 waves sweep disjoint 16-wide column tiles of V, never materializing logits.
// ---------------------------------------------------------------------------
__global__ __launch_bounds__(256) void ctc_gemm_lse_kernel(
    const float* __restrict__ hs,    // (B*T, D)
    const float* __restrict__ W,     // (D, V)
    const float* __restrict__ bias,  // (V)
    float* __restrict__ lse)         // (B*T)
{
    __shared__ float As[16 * 516];        // TDM pads every 256 dw by 2 dw -> stride 516
    __shared__ float red[8][16][2];       // per-wave (max, sumexp) partials per row

    const int tile = blockIdx.x;          // 16000 rows / 16 = 1000 tiles, exact
    const int tid  = threadIdx.x;

    // --- Stage A tile (16x512 f32 = 32KB) via Tensor Data Mover -------------
    if (tid < 32) {
        const unsigned long long ga =
            (unsigned long long)(uintptr_t)(hs + (size_t)tile * 16 * CTC_D);
        const unsigned int lds_off = (unsigned int)(uintptr_t)(void*)As;

        // D# group 0: count=1 | lds_addr | global_addr[56:0] | type=2
        v4u g0;
        g0.x = 1u;                                   // count=1, user mode, no gather
        g0.y = lds_off;                              // LDS byte address of tile start
        g0.z = (unsigned int)(ga & 0xFFFFFFFFull);   // global_addr[31:0]
        g0.w = (unsigned int)((ga >> 32) & 0x01FFFFFFull) | (2u << 30); // [56:32] | type=2

        // D# group 1: 1-D tile of 8192 f32 elements, pad 2 dw per 256 dw stored
        v8i g1;
        g1[0] = (int)((2u << 16)        // data_size = 4 bytes
                    | (1u << 20)        // pad_enable
                    | (7u << 22)        // pad_interval: 256 dwords
                    | (1u << 25));      // pad_amount:   2 dwords
        g1[1] = (int)(( (16u * 512u) & 0xFFFFu) << 16);  // tensor_dim0[15:0]=8192
        g1[2] = (int)(((16u * 512u) >> 16) & 0xFFFFu);   // tensor_dim0[31:16]; tensor_dim1=0
        g1[3] = (int)(((16u * 512u) & 0xFFFFu) << 16);   // tile_dim0 = 8192
        g1[4] = 0;                                       // tile_dim1 = 0 (1-D), tile_dim2 = 0
        g1[5] = (int)(16u * 512u);                       // tensor_dim0_stride = 8192
        g1[6] = 0;
        g1[7] = 0;

        v4i gz = {0, 0, 0, 0};
#if __clang_major__ >= 23
        v8i gz8 = {0, 0, 0, 0, 0, 0, 0, 0};
        __builtin_amdgcn_tensor_load_to_lds(g0, g1, gz, gz, gz8, 0);
#else
        __builtin_amdgcn_tensor_load_to_lds(g0, g1, gz, gz, 0);
#endif
        __builtin_amdgcn_s_wait_tensorcnt(0);
    }
    __syncthreads();

    const int lane = tid & 31;
    const int wave = tid >> 5;
    const int ncol = lane & 15;           // N index of this lane (C/D layout)
    const int kh   = (lane >> 4) << 1;    // K sub-offset for A/B fragments (0 or 2)
    const int mrow = (lane >> 4) << 3;    // C/D row base: lanes 16-31 hold M=8..15

    float runm[8], runs[8];
#pragma unroll
    for (int r = 0; r < 8; ++r) { runm[r] = -3.0e38f; runs[r] = 0.0f; }

    const int VT = (CTC_V + 15) / 16;     // 313 column tiles
    for (int vt = wave; vt < VT; vt += 8) {
        const int v  = vt * 16 + ncol;
        const int vc = (v < CTC_V) ? v : (CTC_V - 1);   // clamp OOB loads
        const float* wp = W + vc;                        // column base, row k at wp[k*V]

        v8f acc = {};
        // K in [0,256): LDS offset = ncol*516 + k
#pragma unroll 4
        for (int k0 = 0; k0 < 256; k0 += 4) {
            v2f a = *(const v2f*)&As[ncol * 516 + k0 + kh];
            v2f bf;
            bf.x = wp[(size_t)(k0 + kh    ) * CTC_V];
            bf.y = wp[(size_t)(k0 + kh + 1) * CTC_V];
            acc = __builtin_amdgcn_wmma_f32_16x16x4_f32(
                false, a, false, bf, (short)0, acc, false, false);
        }
        // K in [256,512): one 2-dword hardware pad crossed -> +2
#pragma unroll 4
        for (int k0 = 256; k0 < 512; k0 += 4) {
            v2f a = *(const v2f*)&As[ncol * 516 + 2 + k0 + kh];
            v2f bf;
            bf.x = wp[(size_t)(k0 + kh    ) * CTC_V];
            bf.y = wp[(size_t)(k0 + kh + 1) * CTC_V];
            acc = __builtin_amdgcn_wmma_f32_16x16x4_f32(
                false, a, false, bf, (short)0, acc, false, false);
        }

        const float bb = bias[vc];
        if (v < CTC_V) {
#pragma unroll
            for (int r = 0; r < 8; ++r) {
                float x  = acc[r] + bb;
                float nm = fmaxf(runm[r], x);
                runs[r] = runs[r] * __expf(runm[r] - nm) + __expf(x - nm);
                runm[r] = nm;
            }
        }
    }

    // Combine across the 16 lanes that share the same M rows (masks stay in half)
#pragma unroll
    for (int m = 1; m < 16; m <<= 1) {
#pragma unroll
        for (int r = 0; r < 8; ++r) {
            float om = __shfl_xor(runm[r], m, 32);
            float os = __shfl_xor(runs[r], m, 32);
            float nm = fmaxf(runm[r], om);
            runs[r] = runs[r] * __expf(runm[r] - nm) + os * __expf(om - nm);
            runm[r] = nm;
        }
    }
    if (ncol == 0) {
#pragma unroll
        for (int r = 0; r < 8; ++r) {
            red[wave][mrow + r][0] = runm[r];
            red[wave][mrow + r][1] = runs[r];
        }
    }
    __syncthreads();

    if (tid < 16) {
        float m = -3.0e38f, s = 0.0f;
        for (int w = 0; w < 8; ++w) {
            float om = red[w][tid][0], os = red[w][tid][1];
            float nm = fmaxf(m, om);
            s = s * __expf(m - nm) + os * __expf(om - nm);
            m = nm;
        }
        lse[(size_t)tile * 16 + tid] = m + __logf(s);
    }
}

// ---------------------------------------------------------------------------
// Kernel B: gather raw logits at the <=101 label classes per batch.
// emitlog[b][j][t] = dot(hs[b,t,:], W[:,c]) + bias[c],  c = j==0 ? blank : ys[b,j-1]
// ---------------------------------------------------------------------------
__global__ __launch_bounds__(256) void ctc_gather_kernel(
    const float* __restrict__ hs,
    const float* __restrict__ W,
    const float* __restrict__ bias,
    const int*   __restrict__ ys_pad,
    float* __restrict__ emitlog)     // (B, J, T)
{
    __shared__ float wc[CTC_D];
    __shared__ float bc;

    const int bj  = blockIdx.x;              // b*J + j
    const int b   = bj / CTC_J;
    const int j   = bj % CTC_J;
    const int tid = threadIdx.x;

    const int c = (j == 0) ? 0 : ys_pad[b * CTC_L + (j - 1)];
    for (int d = tid; d < CTC_D; d += 256) wc[d] = W[(size_t)d * CTC_V + c];
    if (tid == 0) bc = bias[c];
    __syncthreads();

    for (int t = tid; t < CTC_T; t += 256) {
        const float4* hp = (const float4*)(hs + ((size_t)b * CTC_T + t) * CTC_D);
        float acc = 0.0f;
#pragma unroll 8
        for (int d4 = 0; d4 < CTC_D / 4; ++d4) {
            float4 h  = hp[d4];
            float4 w4 = *(const float4*)&wc[d4 * 4];
            acc += h.x * w4.x + h.y * w4.y + h.z * w4.z + h.w * w4.w;
        }
        emitlog[(size_t)bj * CTC_T + t] = acc + bc;
    }
}

// ---------------------------------------------------------------------------
// Kernel C: CTC alpha recursion in log space. One block per batch, one thread
// per extended state s. alpha lives in LDS; two-phase read/write per timestep.
// ---------------------------------------------------------------------------
__global__ __launch_bounds__(256) void ctc_alpha_kernel(
    const float* __restrict__ emitlog,  // (B, J, T)
    const float* __restrict__ lse,      // (B*T)
    const int*   __restrict__ hlens,
    const int*   __restrict__ ys_pad,
    const int*   __restrict__ ys_lens,
    float* __restrict__ out)            // scalar, pre-zeroed
{
    __shared__ float alpha[CTC_S];

    const int b   = blockIdx.x;
    const int tid = threadIdx.x;

    int  jmap = 0;
    bool skip = false;
    if (tid < CTC_S) {
        int ext;
        if (tid & 1) { jmap = (tid >> 1) + 1; ext = ys_pad[b * CTC_L + (tid >> 1)]; }
        else         { jmap = 0;              ext = 0; }
        if (tid >= 2) {
            int extm2 = (tid & 1) ? ys_pad[b * CTC_L + (tid >> 1) - 1] : 0;
            skip = (ext != 0) && (ext != extm2);
        }
    }

    const float* em = emitlog + (size_t)b * CTC_J * CTC_T;
    const float* ls = lse + (size_t)b * CTC_T;

    if (tid < CTC_S) {
        float a0 = NEGV;
        if (tid < 2) a0 = em[(size_t)jmap * CTC_T + 0] - ls[0];
        alpha[tid] = a0;
    }
    __syncthreads();

    const int hl = hlens[b];
    for (int t = 1; t < CTC_T; ++t) {
        float a1 = NEGV, a2 = NEGV, a3 = NEGV, e = 0.0f;
        if (tid < CTC_S) {
            a1 = alpha[tid];
            if (tid >= 1)         a2 = alpha[tid - 1];
            if (tid >= 2 && skip) a3 = alpha[tid - 2];
            e = em[(size_t)jmap * CTC_T + t] - ls[t];
        }
        __syncthreads();
        if (tid < CTC_S && t < hl) {
            float m  = fmaxf(fmaxf(a1, a2), a3);
            alpha[tid] = m + logf(expf(a1 - m) + expf(a2 - m) + expf(a3 - m)) + e;
        }
        __syncthreads();
    }

    if (tid == 0) {
        const int yl = ys_lens[b];
        float l1 = alpha[2 * yl];
        float l2 = alpha[2 * yl - 1];
        float m  = fmaxf(l1, l2);
        float loss = -(m + logf(expf(l1 - m) + expf(l2 - m)));
        if (!(loss < 1e29f)) loss = 0.0f;   // zero_infinity
        atomicAdd(out, loss / (float)CTC_B);
    }
}

__global__ void ctc_zero_out_kernel(float* out) {
    if (threadIdx.x == 0 && blockIdx.x == 0) out[0] = 0.0f;
}

// ---------------------------------------------------------------------------
extern "C" void kernel_launch(void* const* d_in, const int* in_sizes, int n_in,
                              void* d_out, int out_size, void* d_ws, size_t ws_size,
                              hipStream_t stream) {
    const float* hs      = (const float*)d_in[0];   // (B,T,D)
    const int*   hlens   = (const int*)  d_in[1];   // (B)
    const int*   ys_pad  = (const int*)  d_in[2];   // (B,L)
    const int*   ys_lens = (const int*)  d_in[3];   // (B)
    const float* W       = (const float*)d_in[4];   // (D,V)
    const float* bias    = (const float*)d_in[5];   // (V)
    float*       out     = (float*)d_out;

    float* lse     = (float*)d_ws;                          // B*T floats
    float* emitlog = lse + (size_t)CTC_B * CTC_T;           // B*J*T floats
    (void)in_sizes; (void)n_in; (void)out_size; (void)ws_size;

    ctc_zero_out_kernel<<<1, 64, 0, stream>>>(out);
    ctc_gemm_lse_kernel<<<(CTC_B * CTC_T) / 16, 256, 0, stream>>>(hs, W, bias, lse);
    ctc_gather_kernel<<<CTC_B * CTC_J, 256, 0, stream>>>(hs, W, bias, ys_pad, emitlog);
    ctc_alpha_kernel<<<CTC_B, 256, 0, stream>>>(emitlog, lse, hlens, ys_pad, ys_lens, out);
}